// UnarySqrt_54047868453335
// MI455X (gfx1250) — compile-verified
//
#include <hip/hip_runtime.h>
#include <cstdint>
#include <cstddef>

// UnarySqrt scan: out[t] = (1-tr)*x[t] + tr ; tr' = out[t]*(1-tr), per channel.
// Pure streaming problem (536 MB @ 23.3 TB/s ~= 23 us floor). No matmul -> no WMMA.
// CDNA5 path: async global->LDS (ASYNCcnt) depth-4 pipeline, b128, NT load/store policy.

typedef float v4f __attribute__((ext_vector_type(4)));

#define THREADS 256
#define DEPTH   4                      // pipeline stages (power of two)
#define SLOT_BYTES (THREADS * 16)      // 4 KB per stage (one float4 per thread)

#define ASYNC_LOAD_NT(ldsaddr, goff, base64)                                   \
  asm volatile("global_load_async_to_lds_b128 %0, %1, %2 th:TH_LOAD_NT"       \
               :: "v"(ldsaddr), "v"(goff), "s"(base64) : "memory")

// TC = compile-time T (0 => use runtime T)
template <int TC>
__global__ __launch_bounds__(THREADS)
void unary_sqrt_scan(const float* __restrict__ bits,    // [T, N]
                     const float* __restrict__ trace0,  // [N]
                     float* __restrict__ out,           // [T, N]
                     int N, int Trt)
{
  const int T = (TC > 0) ? TC : Trt;

  __shared__ float buf[DEPTH * THREADS * 4];

  const int tid = threadIdx.x;
  const int c4  = blockIdx.x * THREADS + tid;          // float4 channel index
  if (c4 * 4 >= N) return;

  // byte offset of this thread's float4 within one [N] row
  const unsigned row_byte = (unsigned)c4 * 16u;

  // LDS byte address of buf (AS(3) offset) for the async-load destination VGPR
  unsigned lds_base;
  {
    __attribute__((address_space(3))) float* p3 =
        (__attribute__((address_space(3))) float*)buf;
    lds_base = (unsigned)(unsigned long long)p3;
  }
  const unsigned my_lds0 = lds_base + (unsigned)tid * 16u;

  const uint64_t bits_base  = (uint64_t)(uintptr_t)bits;
  const uint64_t row_stride = (uint64_t)N * 4u;

  // ---- prologue: fill DEPTH stages ----
#pragma unroll
  for (int s = 0; s < DEPTH; ++s) {
    if (TC >= DEPTH || s < T) {
      unsigned lds  = my_lds0 + (unsigned)s * SLOT_BYTES;
      uint64_t base = bits_base + (uint64_t)s * row_stride;
      ASYNC_LOAD_NT(lds, row_byte, base);
    }
  }

  v4f tr = *((const v4f*)trace0 + c4);

  int t = 0;
  const int Tmain = (T > DEPTH) ? (T - DEPTH) : 0;

  // ---- steady state: wait oldest stage, consume from LDS, refill slot ----
#pragma unroll DEPTH
  for (; t < Tmain; ++t) {
    asm volatile("s_wait_asynccnt %0" :: "i"(DEPTH - 1) : "memory");
    const int slot = t & (DEPTH - 1);
    v4f x = *reinterpret_cast<const v4f*>(&buf[slot * (THREADS * 4) + tid * 4]);
    // ds_load must land in VGPRs before the async refill can overwrite the slot
    // (DS reads vs async-LDS writes are unordered)
    asm volatile("s_wait_dscnt 0" ::: "memory");
    {
      unsigned lds  = my_lds0 + (unsigned)slot * SLOT_BYTES;
      uint64_t base = bits_base + (uint64_t)(t + DEPTH) * row_stride;
      ASYNC_LOAD_NT(lds, row_byte, base);
    }
    v4f u = 1.0f - tr;
    v4f o = u * x + tr;
    tr = o * u;
    __builtin_nontemporal_store(o, (v4f*)(out + (size_t)t * (size_t)N) + c4);
  }

  // ---- tail: drain remaining in-flight stages ----
  asm volatile("s_wait_asynccnt 0" ::: "memory");
#pragma unroll
  for (int k = 0; k < DEPTH; ++k, ++t) {
    if (TC == 0 && t >= T) break;        // dynamic-T guard only
    const int slot = t & (DEPTH - 1);
    v4f x = *reinterpret_cast<const v4f*>(&buf[slot * (THREADS * 4) + tid * 4]);
    v4f u = 1.0f - tr;
    v4f o = u * x + tr;
    tr = o * u;
    __builtin_nontemporal_store(o, (v4f*)(out + (size_t)t * (size_t)N) + c4);
  }
}

extern "C" void kernel_launch(void* const* d_in, const int* in_sizes, int n_in,
                              void* d_out, int out_size, void* d_ws, size_t ws_size,
                              hipStream_t stream) {
  const float* bits   = (const float*)d_in[0];   // [T, N] float32 {0,1}
  const float* trace0 = (const float*)d_in[1];   // [N] float32
  float* out = (float*)d_out;                    // [T, N] float32

  const int N = in_sizes[1];
  const int T = (N > 0) ? (in_sizes[0] / N) : 0;
  const int blocks = (N + THREADS * 4 - 1) / (THREADS * 4);  // N=1M -> 1024 blocks

  if (T == 64) {
    unary_sqrt_scan<64><<<blocks, THREADS, 0, stream>>>(bits, trace0, out, N, T);
  } else {
    unary_sqrt_scan<0><<<blocks, THREADS, 0, stream>>>(bits, trace0, out, N, T);
  }
}